// AttnPainterOil_27041114095712
// MI455X (gfx1250) — compile-verified
//
#include <hip/hip_runtime.h>

// AttnPainterOil composite: per pixel, select the K highest-indexed strokes
// with alpha > 0.1 and alpha-composite them over a white canvas.
// Memory-latency-bound gather; optimized with batched loads (MLP), wave32
// ballot early-exit, gfx1250 global_prefetch, and loop-carried pointer bumps
// so all intra-batch offsets are 24-bit instruction immediates.

namespace {
constexpr int   kB      = 8;
constexpr int   kN      = 256;
constexpr int   kW      = 128;
constexpr int   kK      = 10;
constexpr int   kWW     = kW * kW;     // 16384 pixels per plane
constexpr float kThresh = 0.1f;
constexpr int   kBatch  = 8;           // alpha planes in flight per iteration
}

__device__ __forceinline__ bool wave_any(bool pred) {
#if __has_builtin(__builtin_amdgcn_ballot_w32)
  return __builtin_amdgcn_ballot_w32(pred) != 0u;   // wave32 on gfx1250
#else
  return __ballot((int)pred) != 0ull;
#endif
}

__global__ __launch_bounds__(256) void attn_painter_composite(
    const float* __restrict__ color,   // [B, N, 3, W, W]
    const float* __restrict__ alpha,   // [B, N, 1, W, W]
    float* __restrict__ out)           // [B, 3, W, W]
{
  const unsigned p   = blockIdx.x * 256u + threadIdx.x;  // pixel id, [0, B*W*W)
  const unsigned b   = p >> 14;                          // p / (W*W)
  const unsigned pix = p & (kWW - 1);                    // h*W + w (lanes -> consecutive w)

  // Base pointers at stroke n = nb (start of the current descending batch).
  // All other offsets inside a batch are compile-time byte constants < 2 MB,
  // i.e. they fit the signed 24-bit VMEM instruction offset.
  const float* aptr = alpha + ((size_t)b * kN * kWW + pix) + (size_t)(kN - kBatch) * kWW;
  const float* cptr = color + ((size_t)b * kN * 3 * kWW + pix) + (size_t)(kN - kBatch) * (3 * kWW);

  float T  = 1.0f;                        // transmittance (front-to-back)
  float C0 = 0.0f, C1 = 0.0f, C2 = 0.0f;  // accumulated premultiplied color
  int   count = 0;

  // Scan strokes from highest id downward in batches of kBatch so the 8
  // independent alpha loads are in flight together (MLP). Wave-uniform early
  // exit once every lane has its K survivors (~2-3 iterations expected).
  for (int nb = kN - kBatch; nb >= 0;
       nb -= kBatch, aptr -= (size_t)kBatch * kWW, cptr -= (size_t)kBatch * (3 * kWW)) {
    float a[kBatch];
#pragma unroll
    for (int j = 0; j < kBatch; ++j)
      a[j] = aptr[(kBatch - 1 - j) * kWW];               // n = nb+7 ... nb (descending)

    if (nb > 0) {
      // Prefetch the next batch of alpha planes (gfx1250 global_prefetch_b8);
      // negative immediate offsets off the same base pointer.
#pragma unroll
      for (int j = 0; j < kBatch; ++j)
        __builtin_prefetch(aptr - (size_t)(j + 1) * kWW, 0, 0);
    }

#pragma unroll
    for (int j = 0; j < kBatch; ++j) {
      if ((count < kK) && (a[j] > kThresh)) {
        const int  noff = (kBatch - 1 - j) * (3 * kWW);  // compile-time constant
        const float w0  = T * a[j];
        C0 = fmaf(w0, cptr[noff],           C0);
        C1 = fmaf(w0, cptr[noff + kWW],     C1);
        C2 = fmaf(w0, cptr[noff + 2 * kWW], C2);
        T *= 1.0f - a[j];
        ++count;
      }
    }
    if (!wave_any(count < kK)) break;
  }

  // Reference-exact rare path: if fewer than K strokes passed the threshold,
  // jax.lax.top_k fills remaining slots with the lowest-index zero-draw
  // strokes, and the reference composites those (behind) with their true
  // alpha. With p(pass)=0.9 over 256 strokes this essentially never fires;
  // the guard is wave-uniform so the common case pays nothing.
  if (wave_any(count < kK)) {
    const float* ap = alpha + ((size_t)b * kN * kWW + pix);
    const float* cp = color + ((size_t)b * kN * 3 * kWW + pix);
    for (int n = 0; n < kN && count < kK; ++n) {
      const float a = ap[(size_t)n * kWW];
      if (a <= kThresh) {
        const float* c = cp + (size_t)n * (3 * kWW);
        const float  w0 = T * a;
        C0 = fmaf(w0, c[0],       C0);
        C1 = fmaf(w0, c[kWW],     C1);
        C2 = fmaf(w0, c[2 * kWW], C2);
        T *= 1.0f - a;
        ++count;
      }
    }
  }

  // Canvas initialized to 1.0 in every channel => add remaining transmittance.
  float* optr = out + ((size_t)b * 3 * kWW + pix);
  optr[0]       = C0 + T;
  optr[kWW]     = C1 + T;
  optr[2 * kWW] = C2 + T;
}

extern "C" void kernel_launch(void* const* d_in, const int* in_sizes, int n_in,
                              void* d_out, int out_size, void* d_ws, size_t ws_size,
                              hipStream_t stream) {
  (void)in_sizes; (void)n_in; (void)d_ws; (void)ws_size; (void)out_size;
  const float* color = (const float*)d_in[0];   // color_stroke [B,N,3,W,W] f32
  const float* alpha = (const float*)d_in[1];   // alpha        [B,N,1,W,W] f32
  float*       out   = (float*)d_out;           // canvas       [B,3,W,W]   f32

  const int total = kB * kWW;                   // one thread per output pixel
  dim3 block(256);
  dim3 grid((unsigned)((total + 255) / 256));
  attn_painter_composite<<<grid, block, 0, stream>>>(color, alpha, out);
}